// DeformableConvLayer_change_17738214932999
// MI455X (gfx1250) — compile-verified
//
#include <hip/hip_runtime.h>

typedef __attribute__((ext_vector_type(2))) float v2f;
typedef __attribute__((ext_vector_type(8))) float v8f;

// Global (address_space(1)) pointer to a float pair: lets us reconstruct the
// address space after storing gather bases as integers in LDS, so the backend
// emits global_load_b64 (not flat_load_b64, which also ties up DScnt/LDS path).
using gv2f_ptr = const v2f __attribute__((address_space(1)))*;

#define BATCH 4
#define HH 256
#define WW 256
#define CC 64
#define TAPS 9
#define KTOT (TAPS * CC)          // 576
#define WAVES_PER_WG 8
#define PIX_PER_WAVE 16

// Zero page (64 floats = 256B): gather target for taps that land in the
// zero-padded border of the feature map.
__global__ void dcn_zero_ws_kernel(float* ws) {
    ws[threadIdx.x] = 0.0f;
}

__global__ __launch_bounds__(256)
void dcn_wmma_kernel(const float* __restrict__ input,
                     const float* __restrict__ gt,
                     const float* __restrict__ offset,
                     const float* __restrict__ kern,
                     const float* __restrict__ bias,
                     float* __restrict__ out,
                     const float* __restrict__ zeropage) {
    __shared__ float kern_lds[KTOT];                                       // 2304 B
    __shared__ unsigned long long addrs[WAVES_PER_WG][TAPS * PIX_PER_WAVE]; // 9216 B

    const int tid  = threadIdx.x;
    const int wave = tid >> 5;
    const int lane = tid & 31;
    const int g    = lane >> 4;   // which K-half this lane feeds (0 or 1)
    const int m    = lane & 15;   // pixel index within wave / B column N

    // Stage kernel weights into LDS (576 f32, cooperative).
    for (int i = tid; i < KTOT; i += 256) kern_lds[i] = kern[i];

    const int pbase = (blockIdx.x * WAVES_PER_WG + wave) * PIX_PER_WAVE;

    // ---- Phase A: offset/mask logic -> per-(pixel,tap) gather base address ----
    for (int it = 0; it < 5; ++it) {
        int task = lane + it * 32;               // 144 tasks over 32 lanes
        if (task < TAPS * PIX_PER_WAVE) {
            int tap = task >> 4;
            int mm  = task & 15;
            int p   = pbase + mm;
            int b   = p >> 16;
            int h   = (p >> 8) & 255;
            int w   = p & 255;
            int ky  = tap / 3, kx = tap % 3;
            int hy  = h + ky - 1, wx = w + kx - 1;
            // extract_image_patches('SAME') on padded index grid: off-image
            // taps read the zero pad -> both coords become 0.
            bool inb = (hy >= 0) && (hy < HH) && (wx >= 0) && (wx < WW);
            int yi = inb ? hy : 0;
            int xi = inb ? wx : 0;

            const float* op = offset + (size_t)p * 18 + tap * 2; // [...,t,{y,x}]
            float y_off = op[0];
            float x_off = op[1];

            // padded mask indexed with UNPADDED coords (reference quirk)
            float pm = (yi >= 1 && xi >= 1) ? gt[(yi - 1) * WW + (xi - 1)] : 0.0f;

            // mask at floored offset point, clipped into padded range [0,257]
            float yof = fminf(fmaxf(floorf((float)yi + y_off), 0.0f), 257.0f);
            float xof = fminf(fmaxf(floorf((float)xi + x_off), 0.0f), 257.0f);
            int yo = (int)yof, xo = (int)xof;
            float pmo = (yo >= 1 && yo <= HH && xo >= 1 && xo <= WW)
                          ? gt[(yo - 1) * WW + (xo - 1)] : 0.0f;
            float diff = (pm != pmo) ? 1.0f : 0.0f;

            float yf = fminf(fmaxf((float)yi + y_off * diff, 0.0f), 255.0f);
            float xf = fminf(fmaxf((float)xi + x_off * diff, 0.0f), 255.0f);
            int y0 = (int)floorf(yf);
            int x0 = (int)floorf(xf);

            // fp[b,y0,x0] indexes the PADDED map: real pixel (y0-1,x0-1),
            // zero row when y0==0 || x0==0.
            const float* basep = (y0 >= 1 && x0 >= 1)
                ? input + (((size_t)b * HH + (y0 - 1)) * WW + (x0 - 1)) * CC
                : zeropage;
            addrs[wave][task] = (unsigned long long)basep;
        }
    }
    __syncthreads();

    // ---- Phase B: D[m][0] = sum_k A[m][k] * kern[k] via 144 fp32 WMMAs ----
    v8f acc = {0.f, 0.f, 0.f, 0.f, 0.f, 0.f, 0.f, 0.f};
    // Only B column N==0 carries the kernel; zero other columns with a
    // branch-free multiply (weights are finite, so x*0 / x*1 is exact).
    const float msk = (m == 0) ? 1.0f : 0.0f;

    for (int tap = 0; tap < TAPS; ++tap) {
        const unsigned long long base = addrs[wave][tap * 16 + m]; // 1 ds_load_b64
        const v2f* ksv = (const v2f*)(kern_lds + tap * CC);        // LDS, pair view
        #pragma unroll
        for (int j = 0; j < 16; ++j) {
            const int koff = 4 * j + 2 * g;        // this lane's K pair
            // A fragment: lane holds A[m][koff], A[m][koff+1]  (16x4 f32 layout)
            v2f a = *(gv2f_ptr)(base + (unsigned long long)(koff * 4));
            // B fragment: uniform ds_load_b64 per half-wave (broadcast), then mask
            v2f kv = ksv[2 * j + g];
            v2f bw = { kv.x * msk, kv.y * msk };
            acc = __builtin_amdgcn_wmma_f32_16x16x4_f32(
                false, a, false, bw, (short)0, acc, false, false);
        }
    }

    // ---- Epilogue: lanes with N==0 (lanes 0,16) hold the 16 results ----
    if (m == 0) {
        float b0 = bias[0];
        float* dst = out + pbase + g * 8;   // lane0 -> pixels 0..7, lane16 -> 8..15
        float4 lo = make_float4(acc[0] + b0, acc[1] + b0, acc[2] + b0, acc[3] + b0);
        float4 hi = make_float4(acc[4] + b0, acc[5] + b0, acc[6] + b0, acc[7] + b0);
        *(float4*)(dst)     = lo;
        *(float4*)(dst + 4) = hi;
    }
}

extern "C" void kernel_launch(void* const* d_in, const int* in_sizes, int n_in,
                              void* d_out, int out_size, void* d_ws, size_t ws_size,
                              hipStream_t stream) {
    const float* input  = (const float*)d_in[0];   // [4,256,256,64] f32
    const float* gt     = (const float*)d_in[1];   // [1,256,256,1] f32
    const float* offset = (const float*)d_in[2];   // [4,256,256,18] f32
    const float* kern   = (const float*)d_in[3];   // [3,3,64,1] f32
    const float* bias   = (const float*)d_in[4];   // [1] f32
    float* out = (float*)d_out;                    // [4,256,256,1] f32
    float* zp  = (float*)d_ws;                     // 256B zero page

    dcn_zero_ws_kernel<<<1, 64, 0, stream>>>(zp);

    const int total_pix = BATCH * HH * WW;                        // 262144
    const int blocks = total_pix / (WAVES_PER_WG * PIX_PER_WAVE); // 2048
    dcn_wmma_kernel<<<blocks, 256, 0, stream>>>(input, gt, offset, kern, bias, out, zp);
}